// ChebConv_11046655885865
// MI455X (gfx1250) — compile-verified
//
#include <hip/hip_runtime.h>
#include <hip/hip_bf16.h>

// ChebConv (K=4) on gfx1250.
// out = x @ (W0 - W2) + Lx @ (W1 + 2*W2 + W3) + bias, Lx = lap2 @ x (SpMM)
// Dense part: V_WMMA_F32_16X16X4_F32 (exact f32), weights staged in LDS,
// B-fragments reused across 5 M-tiles per wave (625 = 125*5, no tails).

typedef __attribute__((ext_vector_type(2))) float v2f;
typedef __attribute__((ext_vector_type(8))) float v8f;

#define N_NODES 10000
#define N_EDGES 160000
#define CH      128          // IN_CH == OUT_CH
#define BATCH   4
#define WSZ     (CH*CH)      // 16384 floats per 128x128 matrix
#define MTILES  5            // M-tiles per block (625 = 125 * 5)

// ---------------- utility: zero a float range ----------------
__global__ void cheb_zero(float* __restrict__ p, int n) {
    int i = blockIdx.x * blockDim.x + threadIdx.x;
    int stride = gridDim.x * blockDim.x;
    for (; i < n; i += stride) p[i] = 0.0f;
}

// ---------------- combine the 4 weight matrices into stacked 256x128 B ----
__global__ void cheb_wcomb(const float* __restrict__ w, float* __restrict__ wc) {
    int i = blockIdx.x * blockDim.x + threadIdx.x;
    if (i < WSZ) {
        float w0 = w[i];
        float w1 = w[WSZ + i];
        float w2 = w[2 * WSZ + i];
        float w3 = w[3 * WSZ + i];
        wc[i]       = w0 - w2;                    // rows 0..127  : Wa
        wc[WSZ + i] = w1 + 2.0f * w2 + w3;        // rows 128..255: Wb
    }
}

// ---------------- degree accumulation ----------------
__global__ void cheb_deg(const long long* __restrict__ ei,
                         const float* __restrict__ ew,
                         float* __restrict__ deg) {
    int e = blockIdx.x * blockDim.x + threadIdx.x;
    if (e < N_EDGES) {
        int r = (int)ei[e];                 // row indices are ei[0..E)
        atomicAdd(&deg[r], ew[e]);
    }
}

// ---------------- deg -> deg^{-1/2} (0 if deg<=0), in place --------------
__global__ void cheb_dinv(float* __restrict__ deg) {
    int n = blockIdx.x * blockDim.x + threadIdx.x;
    if (n < N_NODES) {
        float d = deg[n];
        deg[n] = (d > 0.0f) ? (1.0f / sqrtf(d)) : 0.0f;
    }
}

// ---------------- SpMM scatter: Lx[b, r, f] += coef * x[b, c, f] ---------
// one block per edge, 128 threads = channels, loop over batch
__global__ __launch_bounds__(128)
void cheb_spmm(const float* __restrict__ x,
               const long long* __restrict__ ei,
               const float* __restrict__ ew,
               const float* __restrict__ dinv,
               float* __restrict__ lx) {
    int e = blockIdx.x;
    int f = threadIdx.x;
    int r = (int)ei[e];
    int c = (int)ei[N_EDGES + e];
    // lap2 = 2 * (-dinv[r] * w * dinv[c]); self loops folded into GEMM
    float coef = -2.0f * dinv[r] * ew[e] * dinv[c];
#pragma unroll
    for (int b = 0; b < BATCH; ++b) {
        float v = coef * x[((size_t)b * N_NODES + c) * CH + f];
        atomicAdd(&lx[((size_t)b * N_NODES + r) * CH + f], v);
    }
}

// ---------------- WMMA GEMM: out = [x | (lx - 0.1 x)] @ [Wa ; Wb] + bias --
// block = 256 threads = 8 waves; wave w owns out-channel tile w (16 cols).
// Each block computes 5 consecutive 16-row M-tiles (80 rows) so each LDS
// B-fragment read feeds 5 v_wmma ops. Weights staged in 64KB LDS per phase.
// blockIdx.x = M-group (0..124), blockIdx.y = batch
__global__ __launch_bounds__(256)
void cheb_gemm(const float* __restrict__ x,
               const float* __restrict__ lx,
               const float* __restrict__ wc,     // 256 x 128, row-major
               const float* __restrict__ bias,
               float* __restrict__ out) {
    __shared__ float wlds[WSZ];            // 64 KB: one 128x128 phase

    const int lane = threadIdx.x & 31;
    const int wave = threadIdx.x >> 5;     // n-tile 0..7
    const int m    = lane & 15;            // A row / B-D column within tile
    const int kh   = lane >> 4;            // which K-pair (A), which M-half (D)
    const int rowg = blockIdx.x * (16 * MTILES);
    const int b    = blockIdx.y;
    const int ncol = wave * 16 + m;        // global out channel for B/D

    const float* xr[MTILES];
    const float* lxr[MTILES];
#pragma unroll
    for (int t = 0; t < MTILES; ++t) {
        const size_t roff = ((size_t)b * N_NODES + rowg + 16 * t + m) * CH;
        xr[t]  = x  + roff;
        lxr[t] = lx + roff;
    }

    v8f c[MTILES];
#pragma unroll
    for (int t = 0; t < MTILES; ++t) c[t] = (v8f){};

    // ---- phase 1: stage Wa, compute x @ Wa ----
    {
        const float4* src = (const float4*)wc;          // rows 0..127
        float4* dst = (float4*)wlds;
        for (int i = threadIdx.x; i < WSZ / 4; i += 256) dst[i] = src[i];
    }
    __syncthreads();

#pragma unroll 4
    for (int ks = 0; ks < 32; ++ks) {
        const int k0 = 4 * ks + 2 * kh;    // this lane's K base (per ISA layout)
        v2f bf;
        bf.x = wlds[k0 * CH + ncol];
        bf.y = wlds[(k0 + 1) * CH + ncol];
#pragma unroll
        for (int t = 0; t < MTILES; ++t) {
            v2f a = *(const v2f*)(xr[t] + k0);
            c[t] = __builtin_amdgcn_wmma_f32_16x16x4_f32(
                       false, a, false, bf, (short)0, c[t], false, false);
        }
    }
    __syncthreads();                        // done reading phase-1 LDS

    // ---- phase 2: stage Wb, compute (lx + 2*SELF_LOOP_FILL*x) @ Wb ----
    {
        const float4* src = (const float4*)(wc + WSZ);  // rows 128..255
        float4* dst = (float4*)wlds;
        for (int i = threadIdx.x; i < WSZ / 4; i += 256) dst[i] = src[i];
    }
    __syncthreads();

#pragma unroll 4
    for (int ks = 0; ks < 32; ++ks) {
        const int k0 = 4 * ks + 2 * kh;
        v2f bf;
        bf.x = wlds[k0 * CH + ncol];
        bf.y = wlds[(k0 + 1) * CH + ncol];
#pragma unroll
        for (int t = 0; t < MTILES; ++t) {
            v2f xv = *(const v2f*)(xr[t] + k0);
            v2f lv = *(const v2f*)(lxr[t] + k0);
            v2f a;
            a.x = lv.x - 0.1f * xv.x;       // 2 * (-0.05) = -0.1
            a.y = lv.y - 0.1f * xv.y;
            c[t] = __builtin_amdgcn_wmma_f32_16x16x4_f32(
                       false, a, false, bf, (short)0, c[t], false, false);
        }
    }

    const float bv = bias[ncol];
#pragma unroll
    for (int t = 0; t < MTILES; ++t) {
#pragma unroll
        for (int r = 0; r < 8; ++r) {
            const int row = rowg + 16 * t + r + 8 * kh;   // D layout: M = r + 8*kh
            out[((size_t)b * N_NODES + row) * CH + ncol] = c[t][r] + bv;
        }
    }
}

extern "C" void kernel_launch(void* const* d_in, const int* in_sizes, int n_in,
                              void* d_out, int out_size, void* d_ws, size_t ws_size,
                              hipStream_t stream) {
    const float*     x    = (const float*)d_in[0];       // (B, N, 128)
    const long long* ei   = (const long long*)d_in[1];   // (2, E) int64
    const float*     ew   = (const float*)d_in[2];       // (E,)
    const float*     w    = (const float*)d_in[3];       // (4, 128, 128)
    const float*     bias = (const float*)d_in[4];       // (128,)
    float*           out  = (float*)d_out;               // (B, N, 128)

    // workspace layout (floats): deg/dinv [N] | Wcomb [2*128*128] | Lx [B*N*128]
    float* ws   = (float*)d_ws;
    float* deg  = ws;
    float* wcmb = ws + N_NODES;
    float* lx   = ws + N_NODES + 2 * WSZ;
    const int lx_elems = BATCH * N_NODES * CH;           // 5,120,000 (~20.5 MB)
    (void)ws_size; (void)in_sizes; (void)n_in; (void)out_size;

    // 1) zero accumulators (must happen every call: atomics below)
    cheb_zero<<<256, 256, 0, stream>>>(deg, N_NODES);
    cheb_zero<<<2048, 256, 0, stream>>>(lx, lx_elems);

    // 2) effective weights
    cheb_wcomb<<<(WSZ + 255) / 256, 256, 0, stream>>>(w, wcmb);

    // 3) degree, then deg^{-1/2}
    cheb_deg<<<(N_EDGES + 255) / 256, 256, 0, stream>>>(ei, ew, deg);
    cheb_dinv<<<(N_NODES + 255) / 256, 256, 0, stream>>>(deg);

    // 4) SpMM scatter (edge part of lap2; self loops folded into GEMM)
    cheb_spmm<<<N_EDGES, 128, 0, stream>>>(x, ei, ew, deg, lx);

    // 5) fused dual GEMM + bias via V_WMMA_F32_16X16X4_F32, LDS-staged weights
    dim3 grid(N_NODES / (16 * MTILES), BATCH);   // 125 x 4, exact (no tails)
    cheb_gemm<<<grid, 256, 0, stream>>>(x, lx, wcmb, bias, out);
}